// LS4D_60069412601934
// MI455X (gfx1250) — compile-verified
//
#include <hip/hip_runtime.h>
#include <hip/hip_bf16.h>

typedef __attribute__((ext_vector_type(2))) float v2f;
typedef __attribute__((ext_vector_type(8))) float v8f;

#define BSZ 128
#define LSEQ 2048
#define HD 128
#define TCH 16
#define NCH (LSEQ / TCH)
#define DTC 0.01f
#define SROW 260   // LDS row stride (floats), padded: rows map to distinct bank groups

// ---------------------------------------------------------------------------
// Precompute Ad (complex, 128x128) and W (256x128 real):
//   A = -exp(log_A_real) + i*A_imag ; I = eye broadcast quirk (r==n)
//   denom = 1/(I - A*dt/2); Ad = denom*(I + A*dt/2); Cd = denom*C*dt
//   W[k,m]   = Re(Cd[m,k])          (k <  128)
//   W[k,m]   = -Im(Cd[m,k-128])     (k >= 128)   -> folds the minus sign
// ---------------------------------------------------------------------------
__global__ __launch_bounds__(128) void ls4d_precompute(
    const float* __restrict__ C_ri, const float* __restrict__ logAr,
    const float* __restrict__ Aim, float* __restrict__ AdR,
    float* __restrict__ AdI, float* __restrict__ W)
{
    const int r = blockIdx.x;          // row (also batch index via B==H)
    const int n = threadIdx.x;         // state index
    const int idx = r * HD + n;

    const float ar  = -expf(logAr[idx]);
    const float ai  = Aim[idx];
    const float dt2 = 0.5f * DTC;
    const float Iv  = (r == n) ? 1.0f : 0.0f;

    // denom = 1 / ((Iv - ar*dt2) + i*(-ai*dt2))
    const float dr  = Iv - ar * dt2;
    const float di  = -ai * dt2;
    const float inv = 1.0f / (dr * dr + di * di);
    const float denr =  dr * inv;
    const float deni = -di * inv;

    // Ad = denom * (Iv + a*dt2)
    const float nr = Iv + ar * dt2;
    const float ni = ai * dt2;
    AdR[idx] = denr * nr - deni * ni;
    AdI[idx] = denr * ni + deni * nr;

    // Cd = denom * C * dt
    const float cr = C_ri[2 * idx + 0];
    const float ci = C_ri[2 * idx + 1];
    const float wr = (cr * denr - ci * deni) * DTC;
    const float wi = (cr * deni + ci * denr) * DTC;
    W[n * HD + r]        = wr;     // Re(Cd[r,n]) at W[k=n][m=r]
    W[(HD + n) * HD + r] = -wi;    // -Im(Cd[r,n]) at W[k=128+n][m=r]
}

// ---------------------------------------------------------------------------
// Fused scan + WMMA GEMM.
// grid = B(128) workgroups, 256 threads = 8 wave32.
// Per chunk of 16 timesteps:
//   - threads 0..127 run the elementwise complex recurrence, write the
//     16x256 [Hr|Hi] A-tile to LDS
//   - each wave computes one 16x16 output tile via 64x v_wmma_f32_16x16x4_f32
//     with its 256x16 W-slice resident in 128 VGPRs.
// ---------------------------------------------------------------------------
__global__ __launch_bounds__(256) void ls4d_scan_wmma(
    const float* __restrict__ x, const float* __restrict__ AdR,
    const float* __restrict__ AdI, const float* __restrict__ W,
    float* __restrict__ out)
{
    __shared__ __align__(16) float sH[TCH][SROW];

    const int b     = blockIdx.x;
    const int tid   = threadIdx.x;
    const int lane  = tid & 31;
    const int wv    = tid >> 5;        // wave id 0..7
    const int mbase = wv * 16;         // output-column tile base

    // --- load resident B fragments (W slice for this wave's 16 columns) ---
    // B (4x16) fragment layout mirrors A: lanes 0-15 hold {K0,K1}, lanes
    // 16-31 hold {K2,K3}; lane's column = mbase + (lane&15).
    v2f bfrag[64];
    {
        const int ncol   = mbase + (lane & 15);
        const int kshift = (lane >> 4) << 1;
        #pragma unroll
        for (int kb = 0; kb < 64; ++kb) {
            const int kr = kb * 4 + kshift;
            v2f f;
            f.x = W[kr * HD + ncol];
            f.y = W[(kr + 1) * HD + ncol];
            bfrag[kb] = f;
        }
    }

    // --- per-(b,n) recurrence state, threads 0..127 ---
    float hr = 0.0f, hi = 0.0f, ar = 0.0f, ai = 0.0f;
    if (tid < HD) {
        ar = AdR[b * HD + tid];
        ai = AdI[b * HD + tid];
    }

    const float* __restrict__ xb = x   + (size_t)b * LSEQ * HD;
    float*       __restrict__ ob = out + (size_t)b * LSEQ * HD;

    for (int c = 0; c < NCH; ++c) {
        const int t0 = c * TCH;

        // ---- scan phase: 16 sequential complex steps, deposit A-tile ----
        if (tid < HD) {
            float xv[TCH];
            #pragma unroll
            for (int t = 0; t < TCH; ++t)
                xv[t] = xb[(t0 + t) * HD + tid];   // coalesced per t
            #pragma unroll
            for (int t = 0; t < TCH; ++t) {
                const float nhr = xv[t] + hr * ar - hi * ai;
                const float nhi = hr * ai + hi * ar;
                hr = nhr; hi = nhi;
                sH[t][tid]      = hr;   // k = n       (Hr half)
                sH[t][HD + tid] = hi;   // k = 128 + n (Hi half)
            }
        }
        __syncthreads();

        // ---- WMMA phase: Y(16x16 tile) = [Hr|Hi] (16x256) @ Wslice ----
        v8f acc = {};
        const int mrow   = lane & 15;
        const int kshift = (lane >> 4) << 1;
        #pragma unroll
        for (int kb = 0; kb < 64; ++kb) {
            const v2f af = *(const v2f*)&sH[mrow][kb * 4 + kshift];
            acc = __builtin_amdgcn_wmma_f32_16x16x4_f32(
                false, af, false, bfrag[kb], (short)0, acc, false, false);
        }
        __syncthreads();   // sH reads done before next chunk overwrites

        // ---- store D tile: VGPR j -> row M=j (lanes 0-15) / j+8 (16-31) ----
        const int tt = (lane >> 4) << 3;       // 0 or 8
        const int mm = mbase + (lane & 15);
        #pragma unroll
        for (int j = 0; j < 8; ++j)
            ob[(size_t)(t0 + tt + j) * HD + mm] = acc[j];
    }
}

// ---------------------------------------------------------------------------
extern "C" void kernel_launch(void* const* d_in, const int* in_sizes, int n_in,
                              void* d_out, int out_size, void* d_ws, size_t ws_size,
                              hipStream_t stream) {
    (void)in_sizes; (void)n_in; (void)out_size; (void)ws_size;
    const float* x     = (const float*)d_in[0];   // (B, L, H) f32
    const float* C_ri  = (const float*)d_in[1];   // (H, N2, 2) f32
    const float* logAr = (const float*)d_in[2];   // (H, N2) f32
    const float* Aim   = (const float*)d_in[3];   // (H, N2) f32

    float* ws  = (float*)d_ws;
    float* AdR = ws;                    // 128*128
    float* AdI = ws + HD * HD;          // 128*128
    float* W   = ws + 2 * HD * HD;      // 256*128   (total 256 KB)

    ls4d_precompute<<<HD, HD, 0, stream>>>(C_ri, logAr, Aim, AdR, AdI, W);
    ls4d_scan_wmma<<<BSZ, 256, 0, stream>>>(x, AdR, AdI, W, (float*)d_out);
}